// DataAugment_10316511445759
// MI455X (gfx1250) — compile-verified
//
#include <hip/hip_runtime.h>
#include <hip/hip_bf16.h>

typedef float f4 __attribute__((ext_vector_type(4)));
typedef int v4i __attribute__((__vector_size__(16)));
typedef __attribute__((address_space(1))) v4i as1_v4i;   // global int4
typedef __attribute__((address_space(3))) v4i as3_v4i;   // LDS int4

#if __has_builtin(__builtin_amdgcn_global_load_async_to_lds_b128)
#define HAVE_ASYNC_LDS 1
#else
#define HAVE_ASYNC_LDS 0
#endif

#define CPOL_RT 0   // regular temporal
#define CPOL_NT 1   // TH[2:0] = NT on gfx12-family cpol encoding

#define BATCH      64
#define NPTS       131072
#define CHUNKS     32                       // partial-sum chunks per batch
#define PTS_PER_CHUNK   (NPTS / CHUNKS)     // 4096 points
#define F4_PER_CHUNK    (PTS_PER_CHUNK * 3 / 4)   // 3072 float4
#define RED_THREADS     256
#define F4_PER_THREAD   (F4_PER_CHUNK / RED_THREADS)  // 12
#define XTHREADS        256
#define F4_PER_BLOCK    (XTHREADS * 3)      // 768 float4 per transform block

struct P3 { float x, y, z; };

// ---------------------------------------------------------------------------
// Pass 3 (defined first so the disasm snippet shows it): fused affine
// transform  out = M*p + t + noise, 4 points per thread.
// gfx1250 async path: each wave async-copies its contiguous 1536B tile of
// batch_data and noise global->LDS with three lane-contiguous b128 async
// loads per array (perfectly coalesced, no VGPR round trip), waits
// ASYNCcnt==0 (wave-private => no workgroup barrier), then reads its 48B/lane
// AoS slice from LDS. batch_data RT (L2-resident from pass 1); noise NT so
// the once-read stream doesn't evict batch_data from the 192MB L2.
// Output is NT-stored directly.
// ---------------------------------------------------------------------------
__global__ __launch_bounds__(XTHREADS)
void transform_kernel(const float* __restrict__ data,
                      const int*   __restrict__ flip,
                      const float* __restrict__ scales,
                      const int*   __restrict__ rot,
                      const float* __restrict__ noise,
                      const float* __restrict__ means,
                      float*       __restrict__ out) {
    const int tid = threadIdx.x;
    const size_t g = (size_t)blockIdx.x * XTHREADS + tid;   // 4 points/thread
    const int b = (int)(g >> 15);   // (4*g) / NPTS; uniform per block

    const float sc = scales[b];
    const int   fa = flip[b];
    const float s0 = (fa == 0) ? -sc : sc;
    const float s1 = (fa == 1) ? -sc : sc;
    const float s2 = (fa == 2) ? -sc : sc;

    const float PI4 = 0.78539816339744830962f;
    const float theta = (float)rot[b] * (2.0f * PI4) - PI4; // k*pi/2 - pi/4
    float sn, cs;
    __sincosf(theta, &sn, &cs);

    const float cen0 = s0 * means[b * 3 + 0];
    const float cen2 = s2 * means[b * 3 + 2];
    // effective matrix A = R^T (einsum 'bnd,bde' applies R on the right)
    const float M00 = cs * s0, M02 = -sn * s2;
    const float M20 = sn * s0, M22 =  cs * s2;
    const float M11 = s1;
    const float t0 = cen0 - (cs * cen0 - sn * cen2);
    const float t2 = cen2 - (sn * cen0 + cs * cen2);

    const size_t blockF4 = (size_t)blockIdx.x * F4_PER_BLOCK;
    const size_t vbase   = blockF4 + (size_t)tid * 3;       // this thread's 3 f4
    f4* op = (f4*)out + vbase;

    // speculative far-ahead prefetch of the noise stream (global_prefetch_b8);
    // OOB prefetches are dropped silently per ISA 10.5.
    __builtin_prefetch((const void*)((const f4*)noise + vbase + F4_PER_BLOCK * 8), 0, 0);

    f4 d0, d1, d2, n0, n1, n2;
#if HAVE_ASYNC_LDS
    __shared__ f4 lds_d[F4_PER_BLOCK];
    __shared__ f4 lds_n[F4_PER_BLOCK];
    const int wave = tid >> 5, lane = tid & 31;
    const int wbase = wave * 96;                 // wave's 96-f4 LDS region
    f4* gd = (f4*)((const f4*)data  + blockF4 + wbase + lane);
    f4* gn = (f4*)((const f4*)noise + blockF4 + wbase + lane);
#pragma unroll
    for (int k = 0; k < 3; ++k) {
        __builtin_amdgcn_global_load_async_to_lds_b128(
            (as1_v4i*)(v4i*)(gd + k * 32),
            (as3_v4i*)(v4i*)&lds_d[wbase + lane + k * 32], 0, CPOL_RT);
        __builtin_amdgcn_global_load_async_to_lds_b128(
            (as1_v4i*)(v4i*)(gn + k * 32),
            (as3_v4i*)(v4i*)&lds_n[wbase + lane + k * 32], 0, CPOL_NT);
    }
#if __has_builtin(__builtin_amdgcn_s_wait_asynccnt)
    __builtin_amdgcn_s_wait_asynccnt(0);
#else
    asm volatile("s_wait_asynccnt 0x0" ::: "memory");
#endif
    d0 = lds_d[tid * 3 + 0]; d1 = lds_d[tid * 3 + 1]; d2 = lds_d[tid * 3 + 2];
    n0 = lds_n[tid * 3 + 0]; n1 = lds_n[tid * 3 + 1]; n2 = lds_n[tid * 3 + 2];
#else
    const f4* dp = (const f4*)data  + vbase;
    const f4* np = (const f4*)noise + vbase;
    d0 = dp[0]; d1 = dp[1]; d2 = dp[2];
    n0 = __builtin_nontemporal_load(np + 0);
    n1 = __builtin_nontemporal_load(np + 1);
    n2 = __builtin_nontemporal_load(np + 2);
#endif

    auto pt = [&](float px, float py, float pz,
                  float nx, float ny, float nz) -> P3 {
        P3 o;
        o.x = fmaf(M00, px, fmaf(M02, pz, t0)) + nx;
        o.y = fmaf(M11, py, ny);                      // t_y == 0 exactly
        o.z = fmaf(M20, px, fmaf(M22, pz, t2)) + nz;
        return o;
    };
    const P3 A  = pt(d0.x, d0.y, d0.z,  n0.x, n0.y, n0.z);
    const P3 Bp = pt(d0.w, d1.x, d1.y,  n0.w, n1.x, n1.y);
    const P3 C  = pt(d1.z, d1.w, d2.x,  n1.z, n1.w, n2.x);
    const P3 D  = pt(d2.y, d2.z, d2.w,  n2.y, n2.z, n2.w);

    f4 r0, r1, r2;
    r0.x = A.x;  r0.y = A.y;  r0.z = A.z;  r0.w = Bp.x;
    r1.x = Bp.y; r1.y = Bp.z; r1.z = C.x;  r1.w = C.y;
    r2.x = C.z;  r2.y = D.x;  r2.z = D.y;  r2.w = D.z;

    __builtin_nontemporal_store(r0, op + 0);
    __builtin_nontemporal_store(r1, op + 1);
    __builtin_nontemporal_store(r2, op + 2);
}

// ---------------------------------------------------------------------------
// Pass 1: deterministic per-(batch,chunk) partial sums of raw coordinates.
// Coordinate index of a float4 rotates as (tid + k) % 3; accumulate in a
// rotated frame (compile-time k%3), un-rotate once at the end.
// RT loads on purpose: pulls batch_data into the 192MB L2 so the transform
// pass re-reads it from L2 instead of HBM.
// ---------------------------------------------------------------------------
__global__ __launch_bounds__(RED_THREADS)
void reduce_partial_kernel(const float* __restrict__ data,
                           float* __restrict__ partial /* [B*CHUNKS*3] */) {
    __shared__ float sm[3 * RED_THREADS];
    const int blk   = blockIdx.x;
    const int b     = blk / CHUNKS;
    const int chunk = blk % CHUNKS;
    const int tid   = threadIdx.x;

    const f4* base = (const f4*)(data + (size_t)b * NPTS * 3
                                      + (size_t)chunk * PTS_PER_CHUNK * 3);

    float u0 = 0.f, u1 = 0.f, u2 = 0.f;   // rotated-frame accumulators
#pragma unroll
    for (int k = 0; k < F4_PER_THREAD; ++k) {
        f4 v = base[tid + k * RED_THREADS];
        if (k % 3 == 0)      { u0 += v.x; u1 += v.y; u2 += v.z; u0 += v.w; }
        else if (k % 3 == 1) { u1 += v.x; u2 += v.y; u0 += v.z; u1 += v.w; }
        else                 { u2 += v.x; u0 += v.y; u1 += v.z; u2 += v.w; }
    }
    const int r = tid % 3;
    float a0, a1, a2;
    if (r == 0)      { a0 = u0; a1 = u1; a2 = u2; }
    else if (r == 1) { a0 = u2; a1 = u0; a2 = u1; }
    else             { a0 = u1; a1 = u2; a2 = u0; }

    sm[tid]                   = a0;
    sm[RED_THREADS + tid]     = a1;
    sm[2 * RED_THREADS + tid] = a2;
    for (int off = RED_THREADS / 2; off > 0; off >>= 1) {
        __syncthreads();
        if (tid < off) {
            sm[tid]                   += sm[tid + off];
            sm[RED_THREADS + tid]     += sm[RED_THREADS + tid + off];
            sm[2 * RED_THREADS + tid] += sm[2 * RED_THREADS + tid + off];
        }
    }
    if (tid == 0) {
        partial[blk * 3 + 0] = sm[0];
        partial[blk * 3 + 1] = sm[RED_THREADS];
        partial[blk * 3 + 2] = sm[2 * RED_THREADS];
    }
}

// ---------------------------------------------------------------------------
// Pass 2: reduce CHUNKS partials per (batch, dim) -> mean. Deterministic.
// ---------------------------------------------------------------------------
__global__ void finalize_mean_kernel(const float* __restrict__ partial,
                                     float* __restrict__ means /* [B*3] */) {
    const int t = threadIdx.x;
    if (t < BATCH * 3) {
        const int b = t / 3, d = t % 3;
        float s = 0.f;
        for (int c = 0; c < CHUNKS; ++c)
            s += partial[(b * CHUNKS + c) * 3 + d];
        means[t] = s * (1.0f / (float)NPTS);
    }
}

// ---------------------------------------------------------------------------
// Tail: second tuple element (rotated_gt) value-cast to the fp32 output buffer.
// ---------------------------------------------------------------------------
__global__ void gt_tail_kernel(const int* __restrict__ rot,
                               float* __restrict__ out_tail) {
    const int t = threadIdx.x;
    if (t < BATCH) out_tail[t] = (float)rot[t];
}

extern "C" void kernel_launch(void* const* d_in, const int* in_sizes, int n_in,
                              void* d_out, int out_size, void* d_ws, size_t ws_size,
                              hipStream_t stream) {
    const float* data   = (const float*)d_in[0];  // [64,131072,3] f32
    const int*   flip   = (const int*)  d_in[1];  // [64] i32
    const float* scales = (const float*)d_in[2];  // [64] f32
    const int*   rot    = (const int*)  d_in[3];  // [64] i32
    const float* noise  = (const float*)d_in[4];  // [64,131072,3] f32
    float* out = (float*)d_out;

    float* ws_f    = (float*)d_ws;
    float* means   = ws_f;                 // B*3        = 192 floats
    float* partial = ws_f + BATCH * 3;     // B*CHUNKS*3 = 6144 floats

    (void)in_sizes; (void)n_in; (void)out_size; (void)ws_size;

    reduce_partial_kernel<<<BATCH * CHUNKS, RED_THREADS, 0, stream>>>(data, partial);
    finalize_mean_kernel<<<1, 256, 0, stream>>>(partial, means);

    const int xblocks = (BATCH * NPTS) / 4 / XTHREADS;    // 8192
    transform_kernel<<<xblocks, XTHREADS, 0, stream>>>(
        data, flip, scales, rot, noise, means, out);

    gt_tail_kernel<<<1, 64, 0, stream>>>(rot, out + (size_t)BATCH * NPTS * 3);
}